// ROCKET_69982197121089
// MI455X (gfx1250) — compile-verified
//
#include <hip/hip_runtime.h>
#include <hip/hip_bf16.h>
#include <float.h>
#include <limits.h>

// Problem constants: C_IN=1, SEQ_LEN=512, N_KERNELS=1000, KS_MAX=11, BATCH=64.
#define N_KER  1000
#define BATCH  64
#define SEQL   512
#define KSM    11
#define NROW   (N_KER * KSM)   // 11000 (k,j) rows
#define KPB    4               // kernels per block (one per 64-lane wave pair)
#define TPB    256             // 8 waves: 4 kslots x 64 batch lanes
#define XROW   513             // padded LDS row stride: bank = (b+i)%64
#define SCHUNK 64              // valid-scan chunk length per thread
#define TCH    16              // t-chunk (accumulators held in registers)

#define AS1 __attribute__((address_space(1)))
#define AS3 __attribute__((address_space(3)))

#if defined(__has_builtin)
#if __has_builtin(__builtin_amdgcn_global_load_async_to_lds_b32) && \
    __has_builtin(__builtin_amdgcn_s_wait_asynccnt)
#define USE_ASYNC_STAGE 1
#endif
#endif

// Stage x (64x512 fp32, 128 KB) into LDS rows xs[b*XROW + i].
// Constant trip count (128) -> no exec-mask loop guard. Consecutive lanes
// write consecutive i: coalesced global + conflict-free LDS banks.
__device__ __forceinline__ void stage_x(const float* __restrict__ x, float* xs) {
  int e = (int)threadIdx.x;
#pragma unroll 4
  for (int r = 0; r < (BATCH * SEQL) / TPB; ++r) {
    const int b = e >> 9;
    const int i = e & (SEQL - 1);
#ifdef USE_ASYNC_STAGE
    __builtin_amdgcn_global_load_async_to_lds_b32(
        (AS1 int*)(x + e), (AS3 int*)(xs + b * XROW + i), 0, 0);
#else
    xs[b * XROW + i] = x[e];
#endif
    e += TPB;
  }
#ifdef USE_ASYNC_STAGE
  __builtin_amdgcn_s_wait_asynccnt(0);
#endif
  __syncthreads();
}

// ---------------- preprocessing: first valid t per (k,j) row ----------------
extern "C" __global__ void rocket_init_first(int* __restrict__ first) {
  const int r = (int)blockIdx.x * 256 + (int)threadIdx.x;
  if (r < NROW) first[r] = INT_MAX;
}

extern "C" __global__ void rocket_scan_first(const unsigned char* __restrict__ valid,
                                             int* __restrict__ first,
                                             int maxL, int nchunk) {
  const int gid = (int)blockIdx.x * 256 + (int)threadIdx.x;
  const int row = gid / nchunk;
  const int c   = gid - row * nchunk;
  if (row >= NROW) return;
  const int t0   = c * SCHUNK;
  const int tend = (t0 + SCHUNK < maxL) ? (t0 + SCHUNK) : maxL;
  const unsigned char* v = valid + (size_t)row * maxL;
  for (int t = t0; t < tend; ++t) {
    if (v[t]) { atomicMin(&first[row], t); break; }
  }
}

// ---------------- main kernel, derived-window mode (no idx/valid stream) ----
// Exact reconstruction: valid[row,t] <=> (u32)(t - lo) < 512 with
// lo = first - idx[row, first]; gather index = clip(t-lo,0,511). When the
// window misses, wv==0, so "& 511" (in-bounds junk read) is bit-exact.
extern "C" __global__ void __launch_bounds__(TPB)
rocket_fused_derived(const float* __restrict__ x,
                     const float* __restrict__ weight,
                     const float* __restrict__ bias,
                     const int* __restrict__ idx,
                     const int* __restrict__ l_out,
                     const int* __restrict__ first,
                     float* __restrict__ out,
                     int maxL)
{
  extern __shared__ float xs[];
  stage_x(x, xs);

  const int lane_b = (int)threadIdx.x & 63;
  const int kslot  = (int)threadIdx.x >> 6;
  const int k = __builtin_amdgcn_readfirstlane((int)blockIdx.x * KPB + kslot);
  if (k >= N_KER) return;

  float w[KSM];
  int   lo[KSM];
#pragma unroll
  for (int j = 0; j < KSM; ++j) {
    const int row = k * KSM + j;
    const int tf  = first[row];
    float wj = weight[row];
    int   loj = 0;
    if (tf != INT_MAX) loj = tf - idx[(size_t)row * maxL + tf];
    else               wj = 0.0f;           // no valid position: row is dead
    w[j] = wj;
    lo[j] = loj;
  }
  const float bk = bias[k];
  const int   Lk = __builtin_amdgcn_readfirstlane(l_out[k]);   // >= 1

  const float* __restrict__ xb = xs + lane_b * XROW;
  float mx  = -FLT_MAX;
  int   cnt = 0;

  // ---- full 16-wide t-chunks: accumulators in registers, j-outer ----
  int t0 = 0;
  for (; t0 + TCH <= Lk; t0 += TCH) {
    float acc[TCH];
#pragma unroll
    for (int u = 0; u < TCH; ++u) acc[u] = bk;

#pragma unroll
    for (int j = 0; j < KSM; ++j) {
      const int tb = t0 - lo[j];                 // scalar; window is [0,512)
      if (tb >= 0 && tb <= (SEQL - TCH)) {
        // Fast path: whole chunk inside the window. One base address, then
        // constant dword offsets -> LLVM merges to ds_load_2addr_b32 pairs.
        const float* __restrict__ p = xb + tb;
        const float wj = w[j];
#pragma unroll
        for (int u = 0; u < TCH; ++u) acc[u] = fmaf(wj, p[u], acc[u]);
      } else {
        // Edge path: exact masked form.
#pragma unroll
        for (int u = 0; u < TCH; ++u) {
          const int   tm = tb + u;
          const float wv = ((unsigned)tm < 512u) ? w[j] : 0.0f;
          acc[u] = fmaf(wv, xb[tm & (SEQL - 1)], acc[u]);
        }
      }
    }
#pragma unroll
    for (int u = 0; u < TCH; ++u) {
      mx = fmaxf(mx, acc[u]);
      cnt += (acc[u] > 0.0f) ? 1 : 0;
    }
  }

  // ---- remainder t's (< 16) ----
  for (int t = t0; t < Lk; ++t) {
    float acc = bk;
#pragma unroll
    for (int j = 0; j < KSM; ++j) {
      const int   tm = t - lo[j];
      const float wv = ((unsigned)tm < 512u) ? w[j] : 0.0f;
      acc = fmaf(wv, xb[tm & (SEQL - 1)], acc);
    }
    mx = fmaxf(mx, acc);
    cnt += (acc > 0.0f) ? 1 : 0;
  }

  *(float2*)(out + (size_t)lane_b * (2 * N_KER) + 2 * k) =
      make_float2(mx, (float)cnt / (float)Lk);
}

// ---------------- fallback: direct idx/valid streaming (ws too small) ------
extern "C" __global__ void __launch_bounds__(TPB)
rocket_fused_direct(const float* __restrict__ x,
                    const float* __restrict__ weight,
                    const float* __restrict__ bias,
                    const int* __restrict__ idx,
                    const unsigned char* __restrict__ valid,
                    const int* __restrict__ l_out,
                    float* __restrict__ out,
                    int maxL)
{
  extern __shared__ float xs[];
  stage_x(x, xs);

  const int lane_b = (int)threadIdx.x & 63;
  const int kslot  = (int)threadIdx.x >> 6;
  const int k = __builtin_amdgcn_readfirstlane((int)blockIdx.x * KPB + kslot);
  if (k >= N_KER) return;

  float w[KSM];
#pragma unroll
  for (int j = 0; j < KSM; ++j) w[j] = weight[k * KSM + j];
  const float bk = bias[k];
  const int   Lk = __builtin_amdgcn_readfirstlane(l_out[k]);

  const int*           __restrict__ idxk = idx   + (size_t)k * KSM * (size_t)maxL;
  const unsigned char* __restrict__ valk = valid + (size_t)k * KSM * (size_t)maxL;
  const float*         __restrict__ xb   = xs + lane_b * XROW;

  float mx  = -FLT_MAX;
  int   cnt = 0;

  for (int t0 = 0; t0 < Lk; t0 += 64) {
    const int tend = (t0 + 64 < Lk) ? (t0 + 64) : Lk;
#pragma unroll
    for (int j = 0; j < KSM; ++j) {
      __builtin_prefetch(idxk + (size_t)j * maxL + t0 + 64, 0, 0);
      __builtin_prefetch(valk + (size_t)j * maxL + t0 + 64, 0, 0);
    }
#pragma unroll 4
    for (int t = t0; t < tend; ++t) {
      float acc = bk;
#pragma unroll
      for (int j = 0; j < KSM; ++j) {
        const int   i  = idxk[(size_t)j * maxL + t];
        const float wv = valk[(size_t)j * maxL + t] ? w[j] : 0.0f;
        acc = fmaf(wv, xb[i], acc);
      }
      mx = fmaxf(mx, acc);
      cnt += (acc > 0.0f) ? 1 : 0;
    }
  }

  *(float2*)(out + (size_t)lane_b * (2 * N_KER) + 2 * k) =
      make_float2(mx, (float)cnt / (float)Lk);
}

extern "C" void kernel_launch(void* const* d_in, const int* in_sizes, int n_in,
                              void* d_out, int out_size, void* d_ws, size_t ws_size,
                              hipStream_t stream) {
  const float*         x      = (const float*)d_in[0];
  const float*         weight = (const float*)d_in[1];
  const float*         bias   = (const float*)d_in[2];
  const int*           idx    = (const int*)d_in[3];
  const unsigned char* valid  = (const unsigned char*)d_in[4];
  const int*           l_out  = (const int*)d_in[5];
  float*               out    = (float*)d_out;

  const int maxL = in_sizes[3] / (N_KER * KSM);   // idx is (1000, 11, maxL)

  const dim3   block(TPB);
  const dim3   grid((N_KER + KPB - 1) / KPB);     // 250 blocks
  const size_t shbytes = (size_t)BATCH * XROW * sizeof(float);  // 131,328 B

  if (ws_size >= (size_t)NROW * sizeof(int) && d_ws != nullptr) {
    int* first = (int*)d_ws;
    rocket_init_first<<<dim3((NROW + 255) / 256), block, 0, stream>>>(first);
    const int nchunk = (maxL + SCHUNK - 1) / SCHUNK;
    const int nthr   = NROW * nchunk;
    rocket_scan_first<<<dim3((nthr + 255) / 256), block, 0, stream>>>(valid, first,
                                                                      maxL, nchunk);
    rocket_fused_derived<<<grid, block, shbytes, stream>>>(x, weight, bias, idx,
                                                           l_out, first, out, maxL);
  } else {
    rocket_fused_direct<<<grid, block, shbytes, stream>>>(x, weight, bias, idx,
                                                          valid, l_out, out, maxL);
  }
}